// DifferentiableQuantumCircuit_64209761075286
// MI455X (gfx1250) — compile-verified
//
#include <hip/hip_runtime.h>
#include <math.h>

// ---------------------------------------------------------------------------
// 12-qubit batched statevector sim for MI455X (gfx1250).
// Strategy: one workgroup per sample; 32KB complex state lives in LDS;
// the 12 single-qubit U3 gates per layer are fused into three 16x16 complex
// group gates; each group application is a batched 16x16x16 complex matmul
// done with V_WMMA_F32_16X16X4_F32 chains (full f32 precision).
// HBM traffic is one read of inputs + one write of probabilities (128 MB
// total -> ~5.5us floor at 23.3 TB/s); everything else stays on-chip.
// ---------------------------------------------------------------------------

typedef __attribute__((ext_vector_type(2))) float v2f;
typedef __attribute__((ext_vector_type(8))) float v8f;

#define NUM_QUBITS 12
#define STATE_DIM  4096
#define BATCH      4096
#define NGROUPS    3
#define NLAYERS    2

// --------------------------------------------------------------------------
// Kernel 1: build the six 16x16 complex group gates.
//   G[l][g] = U(4g) (x) U(4g+1) (x) U(4g+2) (x) U(4g+3)   (first factor = MSB)
// One block per (layer,group); thread tid = i*16+j computes G[i][j] as the
// complex product of four 2x2 U3 entries.
// --------------------------------------------------------------------------
__global__ __launch_bounds__(256)
void qsim_build_gates(const float* __restrict__ thetas,
                      const float* __restrict__ phis,
                      const float* __restrict__ lams,
                      float* __restrict__ gr,   // [6][16][16] real
                      float* __restrict__ gi) { // [6][16][16] imag
    const int lg  = blockIdx.x;          // 0..5  = layer*3 + group
    const int l   = lg / NGROUPS;
    const int g   = lg % NGROUPS;
    const int tid = threadIdx.x;         // 0..255
    const int i   = tid >> 4;            // output row
    const int j   = tid & 15;            // output col

    float re = 1.0f, im = 0.0f;
    #pragma unroll
    for (int t = 0; t < 4; ++t) {
        const int q     = g * 4 + t;
        const float th  = thetas[l * NUM_QUBITS + q];
        const float ph  = phis  [l * NUM_QUBITS + q];
        const float la  = lams  [l * NUM_QUBITS + q];
        const float c   = __cosf(0.5f * th);
        const float s   = __sinf(0.5f * th);
        const int ib    = (i >> (3 - t)) & 1;
        const int jb    = (j >> (3 - t)) & 1;
        float er, ei;
        if (ib == 0 && jb == 0) { er = c;                  ei = 0.0f;            }
        else if (ib == 0)        { er = -__cosf(la) * s;    ei = -__sinf(la) * s; }
        else if (jb == 0)        { er =  __cosf(ph) * s;    ei =  __sinf(ph) * s; }
        else { const float pl = ph + la;
               er =  __cosf(pl) * c;    ei =  __sinf(pl) * c; }
        const float nr = re * er - im * ei;
        const float ni = re * ei + im * er;
        re = nr; im = ni;
    }
    gr[lg * 256 + tid] = re;
    gi[lg * 256 + tid] = im;
}

// LDS address of tensor element for a given group sweep.
// State index s = g0*256 + g1*16 + g2 ; contraction axis "col", row id "m",
// tile id "t" (the other two axes packed as t*16+m).
__device__ __forceinline__ int qs_addr(int g, int t, int m, int col) {
    if (g == 2) return t * 256 + m * 16 + col;   // contract bits 3..0 (stride 1)
    if (g == 1) return t * 256 + col * 16 + m;   // contract bits 7..4 (stride 16)
    return col * 256 + t * 16 + m;               // contract bits 11..8 (stride 256)
}

// --------------------------------------------------------------------------
// Kernel 2: one block (256 threads = 8 waves) per sample.
// --------------------------------------------------------------------------
__global__ __launch_bounds__(256)
void qsim_apply(const float* __restrict__ in,   // [B][4096] f32
                const float* __restrict__ gr,   // [6][256]
                const float* __restrict__ gi,   // [6][256]
                float* __restrict__ out) {      // [B][4096] f32
    __shared__ float sr[STATE_DIM];
    __shared__ float si[STATE_DIM];
    __shared__ float red[256];

    const int b    = blockIdx.x;
    const int tid  = threadIdx.x;
    const int lane = tid & 31;
    const int wave = tid >> 5;          // 0..7
    const int half = lane >> 4;         // 0 or 1 (K/M split inside wave)
    const int mrow = lane & 15;

    // ---- load sample, compute L2 norm, normalize into LDS ----
    float acc = 0.0f;
    #pragma unroll
    for (int e = 0; e < 16; ++e) {
        const int idx = e * 256 + tid;
        const float x = in[(size_t)b * STATE_DIM + idx];
        sr[idx] = x;
        acc += x * x;
    }
    red[tid] = acc;
    __syncthreads();
    for (int s = 128; s > 0; s >>= 1) {
        if (tid < s) red[tid] += red[tid + s];
        __syncthreads();
    }
    const float invn = 1.0f / sqrtf(red[0]);
    #pragma unroll
    for (int e = 0; e < 16; ++e) {
        const int idx = e * 256 + tid;
        sr[idx] *= invn;
        si[idx] = 0.0f;
    }

    // ---- 2 layers x 3 group sweeps of 16x16 complex matmuls ----
    for (int l = 0; l < NLAYERS; ++l) {
        for (int g = 0; g < NGROUPS; ++g) {
            __syncthreads();

            // B fragments of W = G^T (shared across tiles/samples, L2-hot).
            // B layout (4x16, row-striped): VGPR0 = rows {4k, 4k+2},
            // VGPR1 = rows {4k+1, 4k+3}; lane&15 = N.
            const float* __restrict__ grp = gr + (l * NGROUPS + g) * 256;
            const float* __restrict__ gip = gi + (l * NGROUPS + g) * 256;
            v2f br[4], bi[4], bn[4];
            #pragma unroll
            for (int k = 0; k < 4; ++k) {
                const int kb = 4 * k + 2 * half;       // B row index
                // W[kb][n] = G[n][kb]
                br[k][0] = grp[mrow * 16 + kb];
                br[k][1] = grp[mrow * 16 + kb + 1];
                bi[k][0] = gip[mrow * 16 + kb];
                bi[k][1] = gip[mrow * 16 + kb + 1];
                bn[k] = -bi[k];
            }

            #pragma unroll
            for (int it = 0; it < 2; ++it) {
                const int t = wave * 2 + it;           // tile 0..15

                // A fragments: lane&15 = M row; half selects K pair.
                v2f ar[4], ai[4];
                #pragma unroll
                for (int k = 0; k < 4; ++k) {
                    const int k0 = 4 * k + 2 * half;
                    const int a0 = qs_addr(g, t, mrow, k0);
                    const int a1 = qs_addr(g, t, mrow, k0 + 1);
                    ar[k][0] = sr[a0]; ar[k][1] = sr[a1];
                    ai[k][0] = si[a0]; ai[k][1] = si[a1];
                }

                // Yr = Xr*Gr^T - Xi*Gi^T ; Yi = Xr*Gi^T + Xi*Gr^T
                v8f cr = {};
                v8f ci = {};
                #pragma unroll
                for (int k = 0; k < 4; ++k)
                    cr = __builtin_amdgcn_wmma_f32_16x16x4_f32(
                        false, ar[k], false, br[k], (short)0, cr, false, false);
                #pragma unroll
                for (int k = 0; k < 4; ++k)
                    cr = __builtin_amdgcn_wmma_f32_16x16x4_f32(
                        false, ai[k], false, bn[k], (short)0, cr, false, false);
                #pragma unroll
                for (int k = 0; k < 4; ++k)
                    ci = __builtin_amdgcn_wmma_f32_16x16x4_f32(
                        false, ar[k], false, bi[k], (short)0, ci, false, false);
                #pragma unroll
                for (int k = 0; k < 4; ++k)
                    ci = __builtin_amdgcn_wmma_f32_16x16x4_f32(
                        false, ai[k], false, br[k], (short)0, ci, false, false);

                // D layout: VGPR v holds (M = v + 8*half, N = lane&15).
                // Tiles are disjoint per wave; same-wave LDS ops are ordered,
                // so in-place read->write is safe; cross-tile mixing only
                // happens at the next sweep (after the barrier above).
                #pragma unroll
                for (int v = 0; v < 8; ++v) {
                    const int mo = v + 8 * half;
                    const int ad = qs_addr(g, t, mo, mrow);
                    sr[ad] = cr[v];
                    si[ad] = ci[v];
                }
            }
        }
    }

    __syncthreads();

    // ---- probabilities ----
    #pragma unroll
    for (int e = 0; e < 16; ++e) {
        const int idx = e * 256 + tid;
        const float xr = sr[idx];
        const float xi = si[idx];
        out[(size_t)b * STATE_DIM + idx] = xr * xr + xi * xi;
    }
}

// --------------------------------------------------------------------------
extern "C" void kernel_launch(void* const* d_in, const int* in_sizes, int n_in,
                              void* d_out, int out_size, void* d_ws, size_t ws_size,
                              hipStream_t stream) {
    const float* inputs = (const float*)d_in[0];  // [4096][4096]
    const float* thetas = (const float*)d_in[1];  // [2][12]
    const float* phis   = (const float*)d_in[2];  // [2][12]
    const float* lams   = (const float*)d_in[3];  // [2][12]
    float* out = (float*)d_out;                   // [4096][4096]

    float* gr = (float*)d_ws;                     // 6*256 floats
    float* gi = gr + NLAYERS * NGROUPS * 256;     // 6*256 floats

    qsim_build_gates<<<NLAYERS * NGROUPS, 256, 0, stream>>>(thetas, phis, lams, gr, gi);
    qsim_apply<<<BATCH, 256, 0, stream>>>(inputs, gr, gi, out);
}